// IndividualLandmarkResNet_19713899889161
// MI455X (gfx1250) — compile-verified
//
#include <hip/hip_runtime.h>

#define CC   3072
#define HWW  784
#define NL   9
#define NCLS 200
#define BB   32

typedef __attribute__((ext_vector_type(2))) float v2f;
typedef __attribute__((ext_vector_type(8))) float v8f;

// d_out layout: all_features (B,C,9) | maps (B,9,784) | scores (B,200,8) | dist (B,9,784)
#define AF_OFF     0
#define MAPS_OFF   (BB * CC * NL)               // 884736
#define SCORES_OFF (MAPS_OFF + BB * NL * HWW)   // 1110528
#define DIST_OFF   (SCORES_OFF + BB * NCLS * 8) // 1161728

#define MSTRIDE 785  // odd -> LDS bank-conflict-free

// ---------------- k0: asq[n] = ||w_land[n]||^2 ----------------
__global__ __launch_bounds__(256) void k0_asq(const float* __restrict__ w_land,
                                              float* __restrict__ asq) {
  __shared__ float red[256];
  const int n = blockIdx.x;          // 0..8
  const int tid = threadIdx.x;
  float s = 0.0f;
  for (int c = tid; c < CC; c += 256) {
    float w = w_land[n * CC + c];
    s = fmaf(w, w, s);
  }
  red[tid] = s;
  __syncthreads();
  for (int off = 128; off > 0; off >>= 1) {
    if (tid < off) red[tid] += red[tid + off];
    __syncthreads();
  }
  if (tid == 0) asq[n] = red[0];
}

// ---------------- k1: dist + maps (pass 1 over feat) ----------------
// grid: 392 blocks x 256 threads. 4 pixel tiles/block, K split across wave pairs.
// LDS: w_land as [c][9] (110 KB) + 8*272 scratch => ~119 KB => 2 workgroups/WGP.
__global__ __launch_bounds__(256) void k1_dist_maps(
    const float* __restrict__ feat, const float* __restrict__ w_land,
    const float* __restrict__ asq, float* __restrict__ maps_out,
    float* __restrict__ dist_out) {
  extern __shared__ float lds[];
  float* wlds    = lds;             // CC*9 floats, layout [c][n]
  float* scratch = lds + CC * NL;   // 8 waves * 272 floats (16x16 tile + 16 bsq)
  const int tid = threadIdx.x;

  // coalesced global reads, conflict-free strided LDS stores (stride 9, gcd(9,64)=1)
  for (int n = 0; n < NL; ++n)
    for (int c = tid; c < CC; c += 256)
      wlds[c * NL + n] = w_land[n * CC + c];
  __syncthreads();

  const int lane   = tid & 31;
  const int wave   = tid >> 5;
  const int tileId = blockIdx.x * 4 + (wave & 3);   // 0..1567
  const int khalf  = wave >> 2;                     // K split: 0 or 1
  const int p0     = tileId << 4;
  const int b      = p0 / HWW;
  const int hw0    = p0 % HWW;                      // tiles never straddle b (16 | 784)
  const int mn     = lane & 15;                     // A: pixel row; B: landmark col
  const int kk     = (lane >> 4) << 1;              // K sub-offset 0 or 2
  const int ns     = (mn < NL) ? mn : 0;            // clamped landmark column
  const float cmsk = (mn < NL) ? 1.0f : 0.0f;       // zero the 7 padded B columns

  const float* fb = feat + (size_t)b * CC * HWW + hw0 + mn;

  v8f acc = {0.f, 0.f, 0.f, 0.f, 0.f, 0.f, 0.f, 0.f};
  float bsq = 0.0f;
  const int cBeg = khalf * (CC / 2);
  const int cEnd = cBeg + (CC / 2);

  for (int kO = cBeg; kO < cEnd; kO += 32) {
    // speculative prefetch ~8 outer iters ahead (OOB tails silently dropped)
    __builtin_prefetch(fb + (size_t)(kO + 256) * HWW, 0, 0);
#pragma unroll
    for (int kb = 0; kb < 32; kb += 4) {
      const int c = kO + kb + kk;
      const float a0 = fb[(size_t)c * HWW];
      const float a1 = fb[(size_t)(c + 1) * HWW];
      v2f a; a[0] = a0; a[1] = a1;
      v2f w;
      w[0] = wlds[c * NL + ns] * cmsk;
      w[1] = wlds[(c + 1) * NL + ns] * cmsk;
      bsq = fmaf(a1, a1, fmaf(a0, a0, bsq));
      acc = __builtin_amdgcn_wmma_f32_16x16x4_f32(
          false, a, false, w, (short)0, acc, false, false);
    }
  }

  // per-pixel ||x||^2: combine the two K sub-lanes within the wave
  bsq += __shfl_xor(bsq, 16);

  float* tb = scratch + wave * 272;
#pragma unroll
  for (int r = 0; r < 8; ++r) {
    const int mm = r + ((lane >> 4) << 3);
    tb[(mm << 4) + (lane & 15)] = acc[r];
  }
  if (lane < 16) tb[256 + lane] = bsq;
  __syncthreads();

  // wave pair (w, w+4) combine; khalf==0 waves run the softmax epilogue
  if (khalf == 0 && lane < 16) {
    const float* tb2 = tb + 4 * 272;
    const int mm = lane;
    const float bs = tb[256 + mm] + tb2[256 + mm];
    float d[NL], e[NL];
    float nmax = -3.402823466e+38f;
#pragma unroll
    for (int n = 0; n < NL; ++n) {
      const float ab = tb[(mm << 4) + n] + tb2[(mm << 4) + n];
      const float dv = bs - 2.0f * ab + asq[n];
      d[n] = dv;
      nmax = fmaxf(nmax, -dv);
    }
    float sum = 0.0f;
#pragma unroll
    for (int n = 0; n < NL; ++n) { e[n] = __expf(-d[n] - nmax); sum += e[n]; }
    const float inv = 1.0f / sum;
    const size_t base = (size_t)b * NL * HWW + hw0 + mm;
#pragma unroll
    for (int n = 0; n < NL; ++n) {
      dist_out[base + (size_t)n * HWW] = d[n];
      maps_out[base + (size_t)n * HWW] = e[n] * inv;
    }
  }
}

// ---------------- k2: all_features = feat @ maps / HW (pass 2 over feat) ----------------
// grid: 32*24 blocks x 256 threads; one 16-channel tile per wave, full K=784.
__global__ __launch_bounds__(256) void k2_pool(
    const float* __restrict__ feat, const float* __restrict__ maps_in,
    float* __restrict__ af_out) {
  extern __shared__ float mlds[];   // 16 * MSTRIDE floats: maps[b] padded to 16 rows
  const int tid = threadIdx.x;
  const int b   = blockIdx.x / 24;
  const int grp = blockIdx.x % 24;

  for (int i = tid; i < 16 * HWW; i += 256) {
    const int n = i / HWW, hw = i - n * HWW;
    mlds[n * MSTRIDE + hw] =
        (n < NL) ? maps_in[((size_t)b * NL + n) * HWW + hw] : 0.0f;
  }
  __syncthreads();

  const int lane  = tid & 31;
  const int wave  = tid >> 5;
  const int cbase = (grp * 8 + wave) << 4;          // 0..3056 step 16
  const int mn    = lane & 15;                      // A: channel row; B: landmark col
  const int kk    = (lane >> 4) << 1;

  const float* frow = feat + ((size_t)b * CC + cbase + mn) * HWW;
  const float* mrow = mlds + mn * MSTRIDE;

  v8f acc = {0.f, 0.f, 0.f, 0.f, 0.f, 0.f, 0.f, 0.f};
  for (int kO = 0; kO < HWW; kO += 16) {
    __builtin_prefetch(frow + kO + 512, 0, 0);      // 2 KB ahead in the row stream
#pragma unroll
    for (int kb = 0; kb < 16; kb += 4) {
      const int k = kO + kb;
      const v2f a = *(const v2f*)(frow + k + kk);   // 8B-aligned contiguous pair
      v2f mv; mv[0] = mrow[k + kk]; mv[1] = mrow[k + kk + 1];
      acc = __builtin_amdgcn_wmma_f32_16x16x4_f32(
          false, a, false, mv, (short)0, acc, false, false);
    }
  }

  if (mn < NL) {
    const float sc = 1.0f / (float)HWW;
#pragma unroll
    for (int r = 0; r < 8; ++r) {
      const int c = cbase + r + ((lane >> 4) << 3);
      af_out[((size_t)b * CC + c) * NL + mn] = acc[r] * sc;
    }
  }
}

// ---------------- k3: scores = w_cls @ (af * modulation), l < 8 ----------------
// grid: 52 blocks x 256 threads; tiles = 32 b * 13 class-tiles.
__global__ __launch_bounds__(256) void k3_scores(
    const float* __restrict__ af, const float* __restrict__ modulation,
    const float* __restrict__ w_cls, float* __restrict__ scores_out) {
  const int tid    = threadIdx.x;
  const int lane   = tid & 31;
  const int wave   = tid >> 5;
  const int tileId = blockIdx.x * 8 + wave;
  if (tileId >= BB * 13) return;
  const int b    = tileId / 13;
  const int mt   = tileId % 13;
  const int cls0 = mt << 4;
  const int mn   = lane & 15;
  const int kk   = (lane >> 4) << 1;

  const int clsA = min(cls0 + mn, NCLS - 1);        // clamp padded rows (never stored)
  const float* wrow = w_cls + (size_t)clsA * CC;
  const int sn = (mn < NL) ? mn : 0;                // dup col 0 for padded cols (never stored)

  v8f acc = {0.f, 0.f, 0.f, 0.f, 0.f, 0.f, 0.f, 0.f};
#pragma unroll 4
  for (int k = 0; k < CC; k += 4) {
    const int c = k + kk;
    const v2f a = *(const v2f*)(wrow + c);
    v2f bv;
    bv[0] = af[((size_t)b * CC + c) * NL + sn] * modulation[c * NL + sn];
    bv[1] = af[((size_t)b * CC + c + 1) * NL + sn] * modulation[(c + 1) * NL + sn];
    acc = __builtin_amdgcn_wmma_f32_16x16x4_f32(
        false, a, false, bv, (short)0, acc, false, false);
  }

  if (mn < 8) {
#pragma unroll
    for (int r = 0; r < 8; ++r) {
      const int kc = cls0 + r + ((lane >> 4) << 3);
      if (kc < NCLS)
        scores_out[((size_t)b * NCLS + kc) * 8 + mn] = acc[r];
    }
  }
}

extern "C" void kernel_launch(void* const* d_in, const int* in_sizes, int n_in,
                              void* d_out, int out_size, void* d_ws, size_t ws_size,
                              hipStream_t stream) {
  const float* feat       = (const float*)d_in[0];
  const float* w_land     = (const float*)d_in[1];
  const float* modulation = (const float*)d_in[2];
  const float* w_cls      = (const float*)d_in[3];

  float* out    = (float*)d_out;
  float* af     = out + AF_OFF;
  float* maps   = out + MAPS_OFF;
  float* scores = out + SCORES_OFF;
  float* dist   = out + DIST_OFF;
  float* asq    = (float*)d_ws;   // 9 floats

  k0_asq<<<NL, 256, 0, stream>>>(w_land, asq);

  const size_t lds1 = (size_t)(CC * NL + 8 * 272) * sizeof(float);  // ~119 KB -> 2 WGs/WGP
  k1_dist_maps<<<(BB * HWW / 16) / 4, 256, lds1, stream>>>(feat, w_land, asq, maps, dist);

  const size_t lds2 = (size_t)(16 * MSTRIDE) * sizeof(float);       // ~50 KB
  k2_pool<<<BB * 24, 256, lds2, stream>>>(feat, maps, af);

  k3_scores<<<(BB * 13 + 7) / 8, 256, 0, stream>>>(af, modulation, w_cls, scores);
}